// PermInvariantSDR_6640019439975
// MI455X (gfx1250) — compile-verified
//
#include <hip/hip_runtime.h>
#include <math.h>

typedef __attribute__((ext_vector_type(2))) float v2f;
typedef __attribute__((ext_vector_type(4))) float v4f;
typedef __attribute__((ext_vector_type(8))) float v8f;

#define T_LEN   131072
#define SLICES  256           // T_LEN/(8*SLICES) = 64 inner iterations per wave
#define NBATCH  32
#define WS_PER_BATCH 24       // [0..3]=Pd, [4..7]=Td, [8+i*4+j]=C[i][j]
#define EPSF 1e-9f

// lexicographic permutations of {0,1,2,3} (itertools.permutations order)
__device__ __constant__ unsigned char PERMS4[24][4] = {
    {0,1,2,3},{0,1,3,2},{0,2,1,3},{0,2,3,1},{0,3,1,2},{0,3,2,1},
    {1,0,2,3},{1,0,3,2},{1,2,0,3},{1,2,3,0},{1,3,0,2},{1,3,2,0},
    {2,0,1,3},{2,0,3,1},{2,1,0,3},{2,1,3,0},{2,3,0,1},{2,3,1,0},
    {3,0,1,2},{3,0,2,1},{3,1,0,2},{3,1,2,0},{3,2,0,1},{3,2,1,0}
};

__global__ void zero_ws_kernel(float* __restrict__ ws) {
    int i = blockIdx.x * blockDim.x + threadIdx.x;
    if (i < NBATCH * WS_PER_BATCH) ws[i] = 0.0f;
}

// One wave (32 threads) per (batch-pair, T-slice).
// Stacked signal matrix S[16][T]: rows 0-3 = pr(b0), 4-7 = t(b0), 8-11 = pr(b1), 12-15 = t(b1).
// Gram chunk via V_WMMA_F32_16X16X4_F32 with A == B (symmetric Gram trick):
// lane L: row s = L&15, K-half h = L>>4 owns timesteps {4h, 4h+1, 4h+2, 4h+3} of each
// 8-step window; float4 load feeds two WMMAs (pairs (x,y) and (z,w)).
__global__ __launch_bounds__(32) void gram_wmma_kernel(const float* __restrict__ pr,
                                                       const float* __restrict__ tt,
                                                       float* __restrict__ ws) {
    const int lane  = threadIdx.x;        // 0..31
    const int s     = lane & 15;          // stacked row
    const int h     = lane >> 4;          // K-half
    const int pairb = blockIdx.x;         // 0..15
    const int b     = 2 * pairb + (s >> 3);
    const int src   = s & 7;              // 0-3: pr, 4-7: t

    const float* row = (src < 4)
        ? pr + ((size_t)b * 4 + src) * (size_t)T_LEN
        : tt + ((size_t)b * 4 + (src - 4)) * (size_t)T_LEN;

    const int   tlen = T_LEN / SLICES;    // 512 (multiple of 8)
    const float* p   = row + (size_t)blockIdx.y * tlen + 4 * h;

    v8f acc0 = {};
    v8f acc1 = {};
    const int nit = tlen >> 3;            // uniform trip count, no divergence
    #pragma unroll 4
    for (int i = 0; i < nit; ++i) {
        v4f v = *(const v4f*)p;           // global_load_b128
        v2f a0 = { v[0], v[1] };
        v2f a1 = { v[2], v[3] };
        // D = A*B + C, A == B == this wave's register pair (symmetric Gram)
        acc0 = __builtin_amdgcn_wmma_f32_16x16x4_f32(
                   false, a0, false, a0, (short)0, acc0, false, false);
        acc1 = __builtin_amdgcn_wmma_f32_16x16x4_f32(
                   false, a1, false, a1, (short)0, acc1, false, false);
        p += 8;
    }
    v8f g = acc0 + acc1;  // lane-wise add of partial Grams is exact

    // Readout: lanes 0..7 own column j=lane of G(b0) (g[r] = G0[r][j]);
    //          lanes 24..31 own column j=lane-24 of G(b1) (g[r] = G1[r][j]).
    int j = -1, ob = 0;
    if (lane < 8)        { j = lane;      ob = 2 * pairb;     }
    else if (lane >= 24) { j = lane - 24; ob = 2 * pairb + 1; }

    if (j >= 0) {
        float* w = ws + ob * WS_PER_BATCH;
        if (j < 4) {
            // diagonal of pr-block -> Pd[j]
            float d = (j == 0) ? g[0] : (j == 1) ? g[1] : (j == 2) ? g[2] : g[3];
            atomicAdd(&w[j], d);
        } else {
            const int jj = j - 4;
            // diagonal of t-block -> Td[jj]
            float d = (j == 4) ? g[4] : (j == 5) ? g[5] : (j == 6) ? g[6] : g[7];
            atomicAdd(&w[4 + jj], d);
            // rows 0..3 of column 4+jj -> C[i][jj] = sum pr_i * t_jj
            atomicAdd(&w[8 + 0 * 4 + jj], g[0]);
            atomicAdd(&w[8 + 1 * 4 + jj], g[1]);
            atomicAdd(&w[8 + 2 * 4 + jj], g[2]);
            atomicAdd(&w[8 + 3 * 4 + jj], g[3]);
        }
    }
}

// One wave: lane b handles batch b; 24-permutation scan + log10, then mean/negate.
__global__ __launch_bounds__(32) void finalize_kernel(const float* __restrict__ ws,
                                                      float* __restrict__ out) {
    const int b = threadIdx.x;  // 0..31
    const float* w = ws + b * WS_PER_BATCH;

    float Pd[4], Td[4], C[4][4];
    #pragma unroll
    for (int i = 0; i < 4; ++i) {
        Pd[i] = w[i];
        Td[i] = w[4 + i];
        #pragma unroll
        for (int j2 = 0; j2 < 4; ++j2) C[i][j2] = w[8 + i * 4 + j2];
    }

    float best = -3.4e38f;
    #pragma unroll 1
    for (int p = 0; p < 24; ++p) {
        float ssum = 0.0f;
        #pragma unroll
        for (int j2 = 0; j2 < 4; ++j2) {
            const int i = PERMS4[p][j2];
            const float e = Pd[i] - 2.0f * C[i][j2] + Td[j2];
            ssum += 10.0f * log10f((Td[j2] + EPSF) / (e + EPSF));
        }
        best = fmaxf(best, ssum * 0.25f);
    }

    // mean over 32 batches (wave32 reduction), negate
    float v = best;
    for (int off = 16; off > 0; off >>= 1) v += __shfl_down(v, off);
    if (b == 0) out[0] = -v * (1.0f / NBATCH);
}

extern "C" void kernel_launch(void* const* d_in, const int* in_sizes, int n_in,
                              void* d_out, int out_size, void* d_ws, size_t ws_size,
                              hipStream_t stream) {
    const float* pr = (const float*)d_in[0];
    const float* tt = (const float*)d_in[1];
    float* ws  = (float*)d_ws;   // needs 32*24*4 = 3 KB
    float* out = (float*)d_out;

    zero_ws_kernel<<<dim3(3), dim3(256), 0, stream>>>(ws);
    gram_wmma_kernel<<<dim3(16, SLICES), dim3(32), 0, stream>>>(pr, tt, ws);
    finalize_kernel<<<dim3(1), dim3(32), 0, stream>>>(ws, out);
}